// TransitionModel_23295902613903
// MI455X (gfx1250) — compile-verified
//
#include <hip/hip_runtime.h>
#include <math.h>
#include <stdint.h>

#define NN 1024   // state dim (square W)
#define BB 128    // batch

typedef __attribute__((ext_vector_type(2))) float v2f;
typedef __attribute__((ext_vector_type(4))) float v4f;
typedef __attribute__((ext_vector_type(8))) float v8f;

// WGP-scope prefetch (scope field = 0 -> "pull into all cache levels on miss").
// __builtin_prefetch only reaches SE/SYS scope (stops at GL2, useless for our
// GL2-resident working set). The 24-bit signed `offset:` immediate lets one
// base VGPR pair serve a whole burst without per-prefetch address VALU.
#define PF_WGP(base, imm)                                                     \
  asm volatile("global_prefetch_b8 %0, off offset:" #imm                      \
               :: "v"((uint64_t)(uintptr_t)(base)))

// ---------------------------------------------------------------------------
// Kernel A1: partial column logsumexp of W over 8 row-chunks of 128 rows.
// grid (NN/256, 8) x 256 threads; thread t handles column k = bx*256+t.
// Fully coalesced: per iteration 256 consecutive floats of one row.
// ---------------------------------------------------------------------------
__global__ __launch_bounds__(256) void k_col_partial(const float* __restrict__ W,
                                                     float* __restrict__ pmax,
                                                     float* __restrict__ psum) {
  int k  = blockIdx.x * 256 + threadIdx.x;
  int r0 = blockIdx.y * (NN / 8);
  float m = -INFINITY, s = 0.f;
  for (int i = 0; i < NN / 8; ++i) {
    float v  = W[(r0 + i) * NN + k];
    float nm = fmaxf(m, v);
    s = s * __expf(m - nm) + __expf(v - nm);   // first iter: exp(-inf)=0
    m = nm;
  }
  pmax[blockIdx.y * NN + k] = m;
  psum[blockIdx.y * NN + k] = s;
}

// Kernel A2: combine the 8 partials into colLSE[k].
__global__ __launch_bounds__(256) void k_col_combine(const float* __restrict__ pmax,
                                                     const float* __restrict__ psum,
                                                     float* __restrict__ colLSE) {
  int k = blockIdx.x * 256 + threadIdx.x;
  float m = -INFINITY;
#pragma unroll
  for (int j = 0; j < 8; ++j) m = fmaxf(m, pmax[j * NN + k]);
  float s = 0.f;
#pragma unroll
  for (int j = 0; j < 8; ++j) s += psum[j * NN + k] * __expf(pmax[j * NN + k] - m);
  colLSE[k] = m + __logf(s);
}

// ---------------------------------------------------------------------------
// Kernel B1: per-row max of W, then E_W[i,k] = exp(W[i,k]-mW[i]).
// One block (256 thr) per row; each thread owns 4 consecutive floats (b128 ld/st).
// ---------------------------------------------------------------------------
__global__ __launch_bounds__(256) void k_row_exp_W(const float* __restrict__ W,
                                                   float* __restrict__ Ew,
                                                   float* __restrict__ mW) {
  __shared__ float red[256];
  int row = blockIdx.x;
  v4f v = *(const v4f*)(W + row * NN + threadIdx.x * 4);
  float m = fmaxf(fmaxf(v.x, v.y), fmaxf(v.z, v.w));
  red[threadIdx.x] = m;
  __syncthreads();
  for (int st = 128; st > 0; st >>= 1) {
    if (threadIdx.x < st) red[threadIdx.x] = fmaxf(red[threadIdx.x], red[threadIdx.x + st]);
    __syncthreads();
  }
  m = red[0];
  v4f e;
  e.x = __expf(v.x - m); e.y = __expf(v.y - m);
  e.z = __expf(v.z - m); e.w = __expf(v.w - m);
  *(v4f*)(Ew + row * NN + threadIdx.x * 4) = e;
  if (threadIdx.x == 0) mW[row] = m;
}

// Kernel B2: g = log_alpha - colLSE, row max, E_g = exp(g - mg).
__global__ __launch_bounds__(256) void k_row_exp_A(const float* __restrict__ LA,
                                                   const float* __restrict__ colLSE,
                                                   float* __restrict__ Eg,
                                                   float* __restrict__ mg) {
  __shared__ float red[256];
  int row = blockIdx.x;
  v4f v = *(const v4f*)(LA + row * NN + threadIdx.x * 4);
  v4f c = *(const v4f*)(colLSE + threadIdx.x * 4);
  v.x -= c.x; v.y -= c.y; v.z -= c.z; v.w -= c.w;
  float m = fmaxf(fmaxf(v.x, v.y), fmaxf(v.z, v.w));
  red[threadIdx.x] = m;
  __syncthreads();
  for (int st = 128; st > 0; st >>= 1) {
    if (threadIdx.x < st) red[threadIdx.x] = fmaxf(red[threadIdx.x], red[threadIdx.x + st]);
    __syncthreads();
  }
  m = red[0];
  v4f e;
  e.x = __expf(v.x - m); e.y = __expf(v.y - m);
  e.z = __expf(v.z - m); e.w = __expf(v.w - m);
  *(v4f*)(Eg + row * NN + threadIdx.x * 4) = e;
  if (threadIdx.x == 0) mg[row] = m;
}

// ---------------------------------------------------------------------------
// Kernel C: fp32 WMMA GEMM  C[b,i] = sum_k E_g[b,k] * E_W[i,k]
// grid 16 x 256 threads (8 wave32s). Block tile: 128 rows (all B) x 64 cols.
// Wave w owns rows [16w,16w+16) x 64 cols -> 4 accumulators of v_wmma_f32_16x16x4_f32.
// A fragment (16x4 f32): lane l<16 holds row r0+l, K=kk..kk+1; lanes 16-31 K=kk+2..kk+3.
// B fragment (4x16 f32): lane l holds B_wmma[k][n=l] = E_W[nb+l][k], same K split.
// K loop runs in 8 panels of 128: panel top prefetches the NEXT panel into the
// WGP cache (scope 0), then a clean unroll-4 inner loop issues b64 loads + wmma.
// Epilogue: out[b,i] = mg[b] + mW[i] + log(C[b,i]).
// ---------------------------------------------------------------------------
__global__ __launch_bounds__(256) void k_wmma_gemm(const float* __restrict__ Eg,
                                                   const float* __restrict__ Ew,
                                                   const float* __restrict__ mg,
                                                   const float* __restrict__ mW,
                                                   float* __restrict__ out) {
  const int lane = threadIdx.x & 31;
  const int wave = threadIdx.x >> 5;
  const int half = lane >> 4;     // K sub-group: 0 -> K0..1, 1 -> K2..3
  const int l    = lane & 15;
  const int r0   = wave * 16;             // batch-row base
  const int nb   = blockIdx.x * 64;       // output-column base

  v8f acc0 = {}, acc1 = {}, acc2 = {}, acc3 = {};

  const float* ap  = Eg + (r0 + l) * NN + 2 * half;
  const float* bp0 = Ew + (nb + 0  + l) * NN + 2 * half;
  const float* bp1 = bp0 + 16 * NN;
  const float* bp2 = bp0 + 32 * NN;
  const float* bp3 = bp0 + 48 * NN;

  for (int kp = 0; kp < NN; kp += 128) {          // 8 K-panels of 128 floats
    if (kp + 128 < NN) {
      // Two computed bases; every other line via the 24-bit offset immediate.
      // Lane-halves are staggered (+64 floats) and each issues 2 lines per
      // stream, jointly covering the full 512 B panel row of every stream.
      const float* an = ap  + kp + 128 + 64 * half;
      const float* bn = bp0 + kp + 128 + 64 * half;
      PF_WGP(an, 0);      PF_WGP(an, 128);
      PF_WGP(bn, 0);      PF_WGP(bn, 128);       // cols nb+0..15
      PF_WGP(bn, 65536);  PF_WGP(bn, 65664);     // cols nb+16..31 (16*NN*4 B)
      PF_WGP(bn, 131072); PF_WGP(bn, 131200);    // cols nb+32..47
      PF_WGP(bn, 196608); PF_WGP(bn, 196736);    // cols nb+48..63
    }
#pragma unroll 4
    for (int kk = kp; kk < kp + 128; kk += 4) {
      v2f a  = *(const v2f*)(ap  + kk);
      v2f b0 = *(const v2f*)(bp0 + kk);
      v2f b1 = *(const v2f*)(bp1 + kk);
      v2f b2 = *(const v2f*)(bp2 + kk);
      v2f b3 = *(const v2f*)(bp3 + kk);
      acc0 = __builtin_amdgcn_wmma_f32_16x16x4_f32(false, a, false, b0, (short)0, acc0, false, false);
      acc1 = __builtin_amdgcn_wmma_f32_16x16x4_f32(false, a, false, b1, (short)0, acc1, false, false);
      acc2 = __builtin_amdgcn_wmma_f32_16x16x4_f32(false, a, false, b2, (short)0, acc2, false, false);
      acc3 = __builtin_amdgcn_wmma_f32_16x16x4_f32(false, a, false, b3, (short)0, acc3, false, false);
    }
  }

  // C/D layout: VGPR v, lane: row = v + 8*half (+r0), col = l (+nb + 16c)
#pragma unroll
  for (int v = 0; v < 8; ++v) {
    int row   = r0 + v + 8 * half;
    float mgb = mg[row];
    out[row * NN + nb + 0  + l] = mgb + mW[nb + 0  + l] + __logf(acc0[v]);
    out[row * NN + nb + 16 + l] = mgb + mW[nb + 16 + l] + __logf(acc1[v]);
    out[row * NN + nb + 32 + l] = mgb + mW[nb + 32 + l] + __logf(acc2[v]);
    out[row * NN + nb + 48 + l] = mgb + mW[nb + 48 + l] + __logf(acc3[v]);
  }
}

// ---------------------------------------------------------------------------
extern "C" void kernel_launch(void* const* d_in, const int* in_sizes, int n_in,
                              void* d_out, int out_size, void* d_ws, size_t ws_size,
                              hipStream_t stream) {
  const float* log_alpha = (const float*)d_in[0];   // (B, N)
  const float* W         = (const float*)d_in[1];   // (N, N)
  float* out = (float*)d_out;                       // (B, N)

  // Workspace layout (floats); total ~4.6 MB.
  float* ws     = (float*)d_ws;
  float* Ew     = ws;                  // NN*NN
  float* Eg     = Ew + NN * NN;        // BB*NN
  float* colLSE = Eg + BB * NN;        // NN
  float* mW     = colLSE + NN;         // NN
  float* mg     = mW + NN;             // BB
  float* pmax   = mg + BB;             // 8*NN
  float* psum   = pmax + 8 * NN;       // 8*NN

  k_col_partial<<<dim3(NN / 256, 8), 256, 0, stream>>>(W, pmax, psum);
  k_col_combine<<<NN / 256, 256, 0, stream>>>(pmax, psum, colLSE);
  k_row_exp_W  <<<NN, 256, 0, stream>>>(W, Ew, mW);
  k_row_exp_A  <<<BB, 256, 0, stream>>>(log_alpha, colLSE, Eg, mg);
  k_wmma_gemm  <<<NN / 64, 256, 0, stream>>>(Eg, Ew, mg, mW, out);
}